// FGMCWAUCHLoss_11089605558409
// MI455X (gfx1250) — compile-verified
//
#include <hip/hip_runtime.h>

// CDNA5 (gfx1250) single-workgroup loss kernel.
// Pairwise hinge with LAMB=1 is linear => collapses to per-column sums.
// Final cross-partial reduction uses V_WMMA_F32_16X16X4_F32 (fp32-exact).

typedef __attribute__((ext_vector_type(2))) float v2f;
typedef __attribute__((ext_vector_type(8))) float v8f;

#define NT    128            // 4 wave32s
#define BATCH 2048
#define NCAT  14
#define NQ    70             // 5 statistics * 14 columns
#define PS    (NT + 1)       // padded LDS stride (bank-conflict free)

__global__ __launch_bounds__(NT)
void fgmcw_auch_loss_kernel(const float* __restrict__ x,
                            const int*   __restrict__ lab,
                            float*       __restrict__ out)
{
    __shared__ float part[NQ * PS];   // 70 * 129 * 4B = 36,120 B
    __shared__ float finals[80];

    const int tid = threadIdx.x;

    // Warm the streaming path (emits global_prefetch_b8).
    __builtin_prefetch(x   + (size_t)tid * NCAT, 0, 0);
    __builtin_prefetch(lab + (size_t)tid * NCAT, 0, 0);

    // ---- Phase 1: per-thread fp32 accumulation of 5 stats x 14 columns ----
    float acc[5][NCAT];
    #pragma unroll
    for (int s = 0; s < 5; ++s)
        #pragma unroll
        for (int c = 0; c < NCAT; ++c) acc[s][c] = 0.f;

    for (int r = tid; r < BATCH; r += NT) {
        const float* xr = x   + r * NCAT;
        const int*   lr = lab + r * NCAT;
        #pragma unroll
        for (int c = 0; c < NCAT; ++c) {
            float xv  = xr[c];
            float lf  = (float)lr[c];
            float p   = 1.f / (1.f + __expf(-xv));   // sigmoid(output)
            float q   = 1.f / (1.f + __expf(-p));    // faithful double sigmoid
            float lq  = __logf(q + 1e-32f);
            float l1q = __logf(1.f - q + 1e-32f);
            acc[0][c] += lf;                 // npos
            acc[1][c] += lf * lq;            // sum lab*log(q+eps)
            acc[2][c] += (1.f - lf) * l1q;   // sum (1-lab)*log(1-q+eps)
            acc[3][c] += lf * p;             // sum p over positives
            acc[4][c] += (1.f - lf) * p;     // sum p over negatives
        }
    }

    #pragma unroll
    for (int s = 0; s < 5; ++s)
        #pragma unroll
        for (int c = 0; c < NCAT; ++c)
            part[(s * NCAT + c) * PS + tid] = acc[s][c];
    __syncthreads();

    // ---- Phase 2: deterministic LDS tree reduction 128 -> 4 partials ----
    for (int s = NT / 2; s >= 4; s >>= 1) {
        if (tid < s)
            for (int q = 0; q < NQ; ++q)
                part[q * PS + tid] += part[q * PS + tid + s];
        __syncthreads();
    }
    // part[q*PS + 0..3] now hold 4 partials per quantity.

    // ---- Phase 3: WMMA f32 16x16x4 folds the 4 K-partials for 16 rows ----
    // A layout (32-bit 16x4): lanes 0-15 = M, VGPR0/1 = K0/K1; lanes 16-31 = K2/K3.
    // B = ones(4x16)  =>  D[m][n] = sum_k A[m][k]  (exact fp32, RNE).
    if (tid < 32) {                       // wave 0: EXEC all-ones inside
        const int lane  = tid;
        const int m     = lane & 15;
        const int kbase = (lane >> 4) * 2;
        v2f bOnes; bOnes.x = 1.f; bOnes.y = 1.f;
        v8f czero = {};
        #pragma unroll
        for (int g = 0; g < 5; ++g) {
            int q = g * 16 + m;
            v2f A; A.x = 0.f; A.y = 0.f;
            if (q < NQ) {
                A.x = part[q * PS + kbase];
                A.y = part[q * PS + kbase + 1];
            }
            v8f d = __builtin_amdgcn_wmma_f32_16x16x4_f32(
                false, A, false, bOnes, (short)0, czero, false, false);
            // D: VGPR r, lanes 0-15 => M=r ; lanes 16-31 => M=r+8
            if (lane == 0) {
                #pragma unroll
                for (int r = 0; r < 8; ++r) {
                    int qq = g * 16 + r;
                    if (qq < NQ) finals[qq] = d[r];
                }
            }
            if (lane == 16) {
                #pragma unroll
                for (int r = 0; r < 8; ++r) {
                    int qq = g * 16 + 8 + r;
                    if (qq < NQ) finals[qq] = d[r];
                }
            }
        }
    }
    __syncthreads();

    // ---- Phase 4: scalar epilogue (14 columns) ----
    if (tid == 0) {
        const float Bf    = (float)BATCH;
        const float total = Bf * (float)NCAT;
        float cel = 0.f, sum_pen = 0.f, last_pen = 0.f;
        for (int c = 0; c < NCAT; ++c) {
            float npos = finals[0 * NCAT + c];
            float slq  = finals[1 * NCAT + c];
            float sl1q = finals[2 * NCAT + c];
            float sp   = finals[3 * NCAT + c];
            float sn   = finals[4 * NCAT + c];
            float nneg = Bf - npos;

            float alpha_P = npos / total;
            float alpha_N = (total - npos) / total;
            float tmp = -alpha_N * (slq / Bf) - alpha_P * (sl1q / Bf);
            cel += ((c == 0) ? 4.f : 1.f) * tmp;

            float pen;
            if (npos == 0.f)      pen = (nneg + sn) / fmaxf(nneg, 1.f);  // no positives
            else if (nneg == 0.f) pen = (npos - sp) / fmaxf(npos, 1.f);  // no negatives
            else                  pen = 1.f - sp / npos + sn / nneg;     // exact linear hinge
            // pen /= LAMB (== 1)
            sum_pen += pen;
            if (c == NCAT - 1) last_pen = pen;
        }
        out[0] = cel + sum_pen / (float)NCAT;  // ALPHA = 1
        out[1] = last_pen;                     // ALPHA * penalties[-1]
    }
}

extern "C" void kernel_launch(void* const* d_in, const int* in_sizes, int n_in,
                              void* d_out, int out_size, void* d_ws, size_t ws_size,
                              hipStream_t stream) {
    (void)in_sizes; (void)n_in; (void)out_size; (void)d_ws; (void)ws_size;
    const float* x   = (const float*)d_in[0];
    const int*   lab = (const int*)d_in[1];
    float*       out = (float*)d_out;
    fgmcw_auch_loss_kernel<<<1, NT, 0, stream>>>(x, lab, out);
}